// MoEFFN_9517647528141
// MI455X (gfx1250) — compile-verified
//
#include <hip/hip_runtime.h>
#include <math.h>

#define NT 4096   // tokens (B*S)
#define NH 1024   // hidden
#define NF 2048   // ffn hidden
#define NE 8      // experts
#define KC 512    // K-chunk staged in LDS (16*512*4 = 32 KB)

typedef __attribute__((ext_vector_type(2))) float v2f;
typedef __attribute__((ext_vector_type(8))) float v8f;
typedef __attribute__((ext_vector_type(4))) unsigned u32x4;
typedef __attribute__((ext_vector_type(8))) unsigned u32x8;

// D = A(16x4, f32) * B(4x16, f32) + C(16x16, f32)   -> v_wmma_f32_16x16x4_f32
__device__ __forceinline__ v8f wmma_f32_k4(v2f a, v2f b, v8f c) {
  return __builtin_amdgcn_wmma_f32_16x16x4_f32(
      /*neg_a=*/false, a, /*neg_b=*/false, b,
      /*c_mod=*/(short)0, c, /*reuse_a=*/false, /*reuse_b=*/false);
}

// Per-lane async copy of 16B global -> LDS, tracked with ASYNCcnt.
__device__ __forceinline__ void async_copy_b128(unsigned lds_byte, const void* g) {
  asm volatile("global_load_async_to_lds_b128 %0, %1, off"
               :: "v"(lds_byte), "v"(g) : "memory");
}
__device__ __forceinline__ void wait_asynccnt0() {
  asm volatile("s_wait_asynccnt 0x0" ::: "memory");
}

__global__ void zero_f32_kernel(float* __restrict__ p, int n) {
  for (int i = blockIdx.x * blockDim.x + threadIdx.x; i < n;
       i += gridDim.x * blockDim.x)
    p[i] = 0.f;
}

// One wave32 per token: router logits, softmax, top-2, p=1 normalize,
// compact (token, weight) into per-expert lists.
__global__ void router_kernel(const float* __restrict__ x,
                              const float* __restrict__ rw,
                              int* __restrict__ counts,
                              int* __restrict__ lists,
                              float* __restrict__ wts) {
  const int lane = threadIdx.x & 31;
  const int wave = threadIdx.x >> 5;
  const int t = blockIdx.x * 8 + wave;
  if (t >= NT) return;

  float acc[NE];
#pragma unroll
  for (int e = 0; e < NE; ++e) acc[e] = 0.f;
  const float* xt = x + (size_t)t * NH;
  for (int h = lane; h < NH; h += 32) {
    float xv = xt[h];
#pragma unroll
    for (int e = 0; e < NE; ++e) acc[e] += xv * rw[e * NH + h];
  }
#pragma unroll
  for (int e = 0; e < NE; ++e) {
    float v = acc[e];
#pragma unroll
    for (int off = 16; off > 0; off >>= 1) v += __shfl_xor(v, off, 32);
    acc[e] = v;
  }
  float mx = acc[0];
#pragma unroll
  for (int e = 1; e < NE; ++e) mx = fmaxf(mx, acc[e]);
  float sum = 0.f;
#pragma unroll
  for (int e = 0; e < NE; ++e) { acc[e] = expf(acc[e] - mx); sum += acc[e]; }
  float inv = 1.f / sum;
#pragma unroll
  for (int e = 0; e < NE; ++e) acc[e] *= inv;
  int e0 = 0; float b0 = acc[0];
#pragma unroll
  for (int e = 1; e < NE; ++e) if (acc[e] > b0) { b0 = acc[e]; e0 = e; }
  int e1 = (e0 == 0) ? 1 : 0; float b1 = acc[e1];
#pragma unroll
  for (int e = 0; e < NE; ++e)
    if (e != e0 && acc[e] > b1) { b1 = acc[e]; e1 = e; }
  float nrm = 1.f / (b0 + b1);
  if (lane == 0) {
    int p0 = atomicAdd(&counts[e0], 1);
    lists[e0 * NT + p0] = t; wts[e0 * NT + p0] = b0 * nrm;
    int p1 = atomicAdd(&counts[e1], 1);
    lists[e1 * NT + p1] = t; wts[e1 * NT + p1] = b1 * nrm;
  }
}

// Stage A (per expert): P[pos, f] = c * silu(x.w1_f) * (x.v1_f) on gathered rows.
// x tile (16 gathered token rows x 512) staged by TDM gather-mode
// tensor_load_to_lds issued by wave 0; tracked with TENSORcnt.
__global__ void moe_ffn1_kernel(const float* __restrict__ x,
                                const float* __restrict__ w1,
                                const float* __restrict__ v1,
                                const int* __restrict__ counts,
                                const int* __restrict__ lists,
                                const float* __restrict__ wts,
                                float* __restrict__ P, int e) {
  __shared__ __align__(16) float xs[16 * KC];
  __shared__ int rows_s[16];
  __shared__ float wrow_s[16];

  const int cnt = counts[e];
  const int m0 = blockIdx.x * 16;
  if (m0 >= cnt) return;

  if (threadIdx.x < 16) {
    int pos = m0 + threadIdx.x;
    rows_s[threadIdx.x] = (pos < cnt) ? lists[e * NT + pos] : 0;  // clamp: masked later
    wrow_s[threadIdx.x] = (pos < cnt) ? wts[e * NT + pos] : 0.f;
  }
  __syncthreads();

  const int lane = threadIdx.x & 31;
  const int wave = threadIdx.x >> 5;
  const int n = lane & 15;
  const int koff = (lane >> 4) << 1;
  const int f0 = blockIdx.y * 128 + wave * 16;

  // Pack the 16 gathered row indices into uniform D# groups 2/3 (16-bit each).
  const int idxv = rows_s[lane & 15];
  u32x4 g2, g3;
#pragma unroll
  for (int w = 0; w < 4; ++w) {
    g2[w] = ((unsigned)__builtin_amdgcn_readlane(idxv, 2 * w) & 0xFFFFu) |
            ((unsigned)__builtin_amdgcn_readlane(idxv, 2 * w + 1) << 16);
    g3[w] = ((unsigned)__builtin_amdgcn_readlane(idxv, 8 + 2 * w) & 0xFFFFu) |
            ((unsigned)__builtin_amdgcn_readlane(idxv, 8 + 2 * w + 1) << 16);
  }
  // D# group 1: wg_mask=0, data_size=4B(2), dim0=KC, dim1=NT, tile=(KC x 16 idx),
  // dim0_stride=NH.
  u32x8 g1;
  g1[0] = 0x00020000u;                 // data_size=2 (4 bytes)
  g1[1] = (KC & 0xFFFFu) << 16;        // tensor_dim0[15:0]
  g1[2] = (KC >> 16) | ((NT & 0xFFFFu) << 16);
  g1[3] = (NT >> 16) | ((KC & 0xFFFFu) << 16);  // tile_dim0 = KC
  g1[4] = 16u;                         // tile_dim1 = 16 valid indices
  g1[5] = (unsigned)NH;                // tensor_dim0_stride[31:0]
  g1[6] = 0u;
  g1[7] = 0u;
  const unsigned lds_base = (unsigned)(uintptr_t)(void*)xs;

  const float* w1p = w1 + (size_t)e * NF * NH + (size_t)(f0 + n) * NH + koff;
  const float* v1p = v1 + (size_t)e * NF * NH + (size_t)(f0 + n) * NH + koff;
  const float* ap = xs + n * KC + koff;

  v8f acc1 = {0.f, 0.f, 0.f, 0.f, 0.f, 0.f, 0.f, 0.f};
  v8f acc2 = {0.f, 0.f, 0.f, 0.f, 0.f, 0.f, 0.f, 0.f};

  for (int kc = 0; kc < NH; kc += KC) {
    if (wave == 0) {
      // D# group 0: count=1, gather_mode=1 (16-bit idx), lds addr, global addr,
      // type=2.
      unsigned long long ga = (unsigned long long)(uintptr_t)(x + kc);
      u32x4 g0;
      g0[0] = 0x80000001u;
      g0[1] = lds_base;
      g0[2] = (unsigned)ga;
      g0[3] = ((unsigned)(ga >> 32) & 0x01FFFFFFu) | 0x80000000u;  // type=2
      asm volatile("tensor_load_to_lds %0, %1, %2, %3"
                   :: "s"(g0), "s"(g1), "s"(g2), "s"(g3) : "memory");
      __builtin_amdgcn_s_wait_tensorcnt(0);
    }
    __syncthreads();
    __builtin_prefetch(w1p + kc + KC, 0, 1);
    __builtin_prefetch(v1p + kc + KC, 0, 1);
    for (int k = 0; k < KC; k += 4) {
      v2f a  = *(const v2f*)(ap + k);
      v2f b1 = *(const v2f*)(w1p + kc + k);
      v2f b2 = *(const v2f*)(v1p + kc + k);
      acc1 = wmma_f32_k4(a, b1, acc1);
      acc2 = wmma_f32_k4(a, b2, acc2);
    }
    __syncthreads();
  }

  const int mb = (lane >> 4) << 3;
#pragma unroll
  for (int j = 0; j < 8; ++j) {
    int M = mb + j;
    if (m0 + M < cnt) {
      float s = acc1[j];
      float hv = (s / (1.f + expf(-s))) * acc2[j] * wrow_s[M];
      P[(size_t)(m0 + M) * NF + f0 + n] = hv;
    }
  }
}

// Stage B (per expert): out[token] += P @ w2[e].
// P tile staged with per-lane global_load_async_to_lds_b128 (ASYNCcnt).
__global__ void moe_ffn2_kernel(const float* __restrict__ P,
                                const float* __restrict__ w2,
                                const int* __restrict__ counts,
                                const int* __restrict__ lists,
                                float* __restrict__ out, int e) {
  __shared__ __align__(16) float ps[16 * KC];
  __shared__ int rows_s[16];

  const int cnt = counts[e];
  const int m0 = blockIdx.x * 16;
  if (m0 >= cnt) return;

  if (threadIdx.x < 16) {
    int pos = m0 + threadIdx.x;
    rows_s[threadIdx.x] = (pos < cnt) ? lists[e * NT + pos] : 0;
  }
  __syncthreads();

  const int lane = threadIdx.x & 31;
  const int wave = threadIdx.x >> 5;
  const int n = lane & 15;
  const int koff = (lane >> 4) << 1;
  const int h0 = blockIdx.y * 128 + wave * 16;

  const float* bp = w2 + (size_t)e * NF * NH + (size_t)koff * NH + h0 + n;
  const float* ap = ps + n * KC + koff;

  v8f acc = {0.f, 0.f, 0.f, 0.f, 0.f, 0.f, 0.f, 0.f};

  for (int kc = 0; kc < NF; kc += KC) {
    // async-stage P chunk (rows beyond cnt are garbage but masked at the store)
    for (int i = threadIdx.x; i < 16 * (KC / 4); i += 256) {
      int r = i >> 7;                 // KC/4 == 128 b128 units per row
      int c4 = (i & 127) << 2;
      unsigned lds_byte = (unsigned)(uintptr_t)(void*)(ps + r * KC + c4);
      async_copy_b128(lds_byte, P + (size_t)(m0 + r) * NF + kc + c4);
    }
    wait_asynccnt0();
    __syncthreads();
    for (int k = 0; k < KC; k += 4) {
      v2f a = *(const v2f*)(ap + k);
      v2f b;
      b.x = bp[(size_t)(kc + k) * NH];
      b.y = bp[(size_t)(kc + k) * NH + NH];
      acc = wmma_f32_k4(a, b, acc);
    }
    __syncthreads();
  }

  const int mb = (lane >> 4) << 3;
#pragma unroll
  for (int j = 0; j < 8; ++j) {
    int M = mb + j;
    if (m0 + M < cnt) {
      int t = rows_s[M];
      out[(size_t)t * NH + h0 + n] += acc[j];
    }
  }
}

extern "C" void kernel_launch(void* const* d_in, const int* in_sizes, int n_in,
                              void* d_out, int out_size, void* d_ws, size_t ws_size,
                              hipStream_t stream) {
  const float* x  = (const float*)d_in[0];
  const float* rw = (const float*)d_in[1];
  const float* w1 = (const float*)d_in[2];
  const float* v1 = (const float*)d_in[3];
  const float* w2 = (const float*)d_in[4];
  float* out = (float*)d_out;

  char* ws = (char*)d_ws;
  int*   counts = (int*)ws;                              // 8 ints
  int*   lists  = (int*)(ws + 1024);                     // NE*NT ints (128 KB)
  float* wts    = (float*)(ws + 1024 + NE * NT * 4);     // NE*NT floats (128 KB)
  float* P      = (float*)(ws + 1024 + 2 * NE * NT * 4); // NT*NF floats (32 MB)

  zero_f32_kernel<<<1, 64, 0, stream>>>((float*)counts, NE);
  zero_f32_kernel<<<1024, 256, 0, stream>>>(out, out_size);
  router_kernel<<<NT / 8, 256, 0, stream>>>(x, rw, counts, lists, wts);

  for (int e = 0; e < NE; ++e) {
    moe_ffn1_kernel<<<dim3(NT / 16, NF / 128), 256, 0, stream>>>(
        x, w1, v1, counts, lists, wts, P, e);
    moe_ffn2_kernel<<<dim3(NT / 16, NH / 128), 256, 0, stream>>>(
        P, w2, counts, lists, out, e);
  }
}